// GCN_63556926046908
// MI455X (gfx1250) — compile-verified
//
#include <hip/hip_runtime.h>
#include <cstdint>

typedef float v2f __attribute__((ext_vector_type(2)));
typedef float v8f __attribute__((ext_vector_type(8)));

#define F_DIM 100   // raw features
#define H_DIM 16    // hidden
#define C_DIM 40    // classes
#define CP_DIM 48   // padded classes (3 x 16 WMMA tiles)
#define D_DIM 500   // embedded dim
#define EPSV 1e-5f

// ---------------- init: deg=1 (self loop), zero stats accumulators ----------
__global__ void init_k(float* __restrict__ deg, float* __restrict__ sums, int n) {
    int i = blockIdx.x * blockDim.x + threadIdx.x;
    if (i < n) deg[i] = 1.0f;
    if (i < 256) sums[i] = 0.0f;
}

// ---------------- per-feature column sums / sumsq of x ----------------------
__global__ __launch_bounds__(256) void colstats_k(const float* __restrict__ x,
                                                  float* __restrict__ sum,
                                                  float* __restrict__ sumsq, int n) {
    __shared__ float ss[F_DIM];
    __shared__ float sq[F_DIM];
    int t = threadIdx.x;
    for (int j = t; j < F_DIM; j += 256) { ss[j] = 0.f; sq[j] = 0.f; }
    __syncthreads();
    int r = blockIdx.x * 256 + t;
    if (r < n) {
        const float* xr = x + (size_t)r * F_DIM;
        for (int jj = 0; jj < F_DIM; ++jj) {
            int j = jj + t; j -= (j >= F_DIM) ? F_DIM : 0; j -= (j >= F_DIM) ? F_DIM : 0;
            j %= F_DIM;  // stagger to spread LDS-atomic conflicts
            float v = xr[j];
            atomicAdd(&ss[j], v);
            atomicAdd(&sq[j], v * v);
        }
    }
    __syncthreads();
    for (int j = t; j < F_DIM; j += 256) {
        atomicAdd(&sum[j], ss[j]);
        atomicAdd(&sumsq[j], sq[j]);
    }
}

// ---------------- in-degree accumulation over dst ---------------------------
__global__ void deg_k(const long long* __restrict__ dst, float* __restrict__ deg, int e) {
    int i = blockIdx.x * blockDim.x + threadIdx.x;
    if (i < e) atomicAdd(&deg[(int)dst[i]], 1.0f);
}

// ---------------- deg -> rsqrt(deg), in place --------------------------------
__global__ void dinv_k(float* __restrict__ d, int n) {
    int i = blockIdx.x * blockDim.x + threadIdx.x;
    if (i < n) d[i] = rsqrtf(d[i]);
}

// ---------------- fold embed+BN into Weff/cshift, pad W2 --------------------
__global__ __launch_bounds__(256) void params_k(
    const float* __restrict__ sum, const float* __restrict__ sumsq,
    const float* __restrict__ fe, const float* __restrict__ ve,
    const float* __restrict__ gamma, const float* __restrict__ beta,
    const float* __restrict__ W1, const float* __restrict__ W2,
    float* __restrict__ Weff, float* __restrict__ cshift,
    float* __restrict__ W2pad, float invN)
{
    __shared__ float sscale[D_DIM];
    __shared__ float sshift[D_DIM];
    int t = threadIdx.x;
    for (int d = t; d < D_DIM; d += 256) {
        int j = d / 5, k = d % 5;
        float e  = (k < 4) ? fe[j * 4 + k] : ve[j];
        float mean = sum[j] * invN;
        float var  = sumsq[j] * invN - mean * mean;
        float r  = rsqrtf(var * e * e + EPSV);
        float g  = gamma[d];
        sscale[d] = g * e * r;
        sshift[d] = beta[d] - mean * e * r * g;
    }
    __syncthreads();
    for (int i = t; i < F_DIM * H_DIM; i += 256) {
        int j = i / H_DIM, m = i % H_DIM;
        float acc = 0.f;
        #pragma unroll
        for (int k = 0; k < 5; ++k)
            acc += sscale[j * 5 + k] * W1[(j * 5 + k) * H_DIM + m];
        Weff[i] = acc;
    }
    if (t < H_DIM) {
        float acc = 0.f;
        for (int d = 0; d < D_DIM; ++d) acc += sshift[d] * W1[d * H_DIM + t];
        cshift[t] = acc;
    }
    for (int i = t; i < H_DIM * CP_DIM; i += 256) {
        int k = i / CP_DIM, c = i % CP_DIM;
        W2pad[i] = (c < C_DIM) ? W2[k * C_DIM + c] : 0.f;
    }
}

// ---------------- GEMM1: hpre = x @ Weff + cshift  (fp32 WMMA, K=100) -------
__global__ __launch_bounds__(256) void gemm1_wmma_k(
    const float* __restrict__ x, const float* __restrict__ Weff,
    const float* __restrict__ cshift, float* __restrict__ hpre, int nrows)
{
    int wave = threadIdx.x >> 5;
    int lane = threadIdx.x & 31;
    int tile = blockIdx.x * 8 + wave;
    int ntiles = (nrows + 15) >> 4;
    if (tile >= ntiles) return;          // wave-uniform: EXEC all-ones for WMMA
    int r0 = tile << 4;
    int half = lane >> 4;                // lanes 16-31 carry K+2 for fp32 A/B
    int n = lane & 15;
    int arow = r0 + n; if (arow >= nrows) arow = nrows - 1;
    const float* xr = x + (size_t)arow * F_DIM;
    v8f c = {};
    for (int k0 = 0; k0 < F_DIM; k0 += 4) {
        int kk = k0 + half * 2;
        v2f a, b;
        a.x = xr[kk];
        a.y = xr[kk + 1];
        b.x = Weff[kk * H_DIM + n];
        b.y = Weff[(kk + 1) * H_DIM + n];
        c = __builtin_amdgcn_wmma_f32_16x16x4_f32(false, a, false, b,
                                                  (short)0, c, false, false);
    }
    float cs = cshift[n];
    #pragma unroll
    for (int r = 0; r < 8; ++r) {
        int row = r0 + r + half * 8;
        if (row < nrows) hpre[(size_t)row * H_DIM + n] = c[r] + cs;
    }
}

// ---------------- self-loop init: agg = dinv^2 * h --------------------------
__global__ void agg_init_k(const float* __restrict__ dinv, const float* __restrict__ h,
                           float* __restrict__ agg, int total) {
    int i = blockIdx.x * blockDim.x + threadIdx.x;
    if (i >= total) return;
    float dv = dinv[i >> 4];
    agg[i] = dv * dv * h[i];
}

// ---------------- edge scatter: 16 lanes per edge, L2 f32 atomics -----------
__global__ __launch_bounds__(256) void scatter_k(
    const long long* __restrict__ src, const long long* __restrict__ dst,
    const float* __restrict__ dinv, const float* __restrict__ h,
    float* __restrict__ agg, int e)
{
    int idx = blockIdx.x * blockDim.x + threadIdx.x;
    int ei = idx >> 4;
    int m  = idx & 15;
    if (ei >= e) return;
    int s = (int)src[ei];
    int d = (int)dst[ei];
    float nrm = dinv[s] * dinv[d];
    atomicAdd(&agg[(size_t)d * H_DIM + m], nrm * h[(size_t)s * H_DIM + m]);
}

// ---------------- h1 = relu(agg1 + b1) in place; agg2 = dinv^2 * h1 ---------
__global__ void relu_agg_k(float* __restrict__ h1, const float* __restrict__ b1,
                           const float* __restrict__ dinv, float* __restrict__ agg2,
                           int total) {
    int i = blockIdx.x * blockDim.x + threadIdx.x;
    if (i >= total) return;
    float v = h1[i] + b1[i & 15];
    v = fmaxf(v, 0.f);
    h1[i] = v;
    float dv = dinv[i >> 4];
    agg2[i] = dv * dv * v;
}

// ---------------- GEMM2: out = agg2 @ W2pad + b2  (fp32 WMMA, K=16) ---------
__global__ __launch_bounds__(256) void gemm2_wmma_k(
    const float* __restrict__ A, const float* __restrict__ W2pad,
    const float* __restrict__ b2, float* __restrict__ out, int nrows)
{
    int wave = threadIdx.x >> 5;
    int lane = threadIdx.x & 31;
    int tile = blockIdx.x * 8 + wave;
    int ntiles = (nrows + 15) >> 4;
    if (tile >= ntiles) return;
    int r0 = tile << 4;
    int half = lane >> 4;
    int n = lane & 15;
    int arow = r0 + n; if (arow >= nrows) arow = nrows - 1;
    const float* ar = A + (size_t)arow * H_DIM;
    v8f cc[3] = {{}, {}, {}};
    #pragma unroll
    for (int k0 = 0; k0 < H_DIM; k0 += 4) {
        int kk = k0 + half * 2;
        v2f a;
        a.x = ar[kk];
        a.y = ar[kk + 1];
        const float* w0 = W2pad + kk * CP_DIM + n;
        const float* w1 = W2pad + (kk + 1) * CP_DIM + n;
        #pragma unroll
        for (int tcol = 0; tcol < 3; ++tcol) {
            v2f b;
            b.x = w0[tcol * 16];
            b.y = w1[tcol * 16];
            cc[tcol] = __builtin_amdgcn_wmma_f32_16x16x4_f32(
                false, a, false, b, (short)0, cc[tcol], false, false);
        }
    }
    #pragma unroll
    for (int tcol = 0; tcol < 3; ++tcol) {
        int col = tcol * 16 + n;
        if (col < C_DIM) {
            float bb = b2[col];
            #pragma unroll
            for (int r = 0; r < 8; ++r) {
                int row = r0 + r + half * 8;
                if (row < nrows) out[(size_t)row * C_DIM + col] = cc[tcol][r] + bb;
            }
        }
    }
}

// ---------------- in-place row log_softmax over 40 classes ------------------
__global__ void logsoftmax_k(float* __restrict__ out, int n) {
    int r = blockIdx.x * blockDim.x + threadIdx.x;
    if (r >= n) return;
    float* p = out + (size_t)r * C_DIM;
    float v[C_DIM];
    float mx = -3.4e38f;
    #pragma unroll 8
    for (int c = 0; c < C_DIM; ++c) { v[c] = p[c]; mx = fmaxf(mx, v[c]); }
    float s = 0.f;
    #pragma unroll 8
    for (int c = 0; c < C_DIM; ++c) s += expf(v[c] - mx);
    float l = logf(s) + mx;
    #pragma unroll 8
    for (int c = 0; c < C_DIM; ++c) p[c] = v[c] - l;
}

extern "C" void kernel_launch(void* const* d_in, const int* in_sizes, int n_in,
                              void* d_out, int out_size, void* d_ws, size_t ws_size,
                              hipStream_t stream) {
    (void)n_in; (void)out_size; (void)ws_size;
    const float*     x     = (const float*)d_in[0];
    const long long* ei    = (const long long*)d_in[1];   // int64 [2, E]
    const float*     fe    = (const float*)d_in[2];
    const float*     ve    = (const float*)d_in[3];
    const float*     gamma = (const float*)d_in[4];
    const float*     beta  = (const float*)d_in[5];
    const float*     W1    = (const float*)d_in[6];
    const float*     b1    = (const float*)d_in[7];
    const float*     W2    = (const float*)d_in[8];
    const float*     b2    = (const float*)d_in[9];
    float* out = (float*)d_out;

    int n = in_sizes[0] / F_DIM;
    int e = in_sizes[1] / 2;

    // workspace layout (floats)
    float* ws     = (float*)d_ws;
    float* sums   = ws;             // [256] = sum[128] | sumsq[128]
    float* sumx   = ws;
    float* sumsq  = ws + 128;
    float* Weff   = ws + 256;       // [100*16]
    float* cshift = ws + 1856;      // [16]
    float* W2pad  = ws + 1872;      // [16*48]
    size_t o = 2688;
    float* dinv = ws + o; o += (size_t)((n + 15) & ~15);
    float* hpre = ws + o; o += (size_t)n * H_DIM;
    float* agg1 = ws + o; o += (size_t)n * H_DIM;   // later reused as h1 (in place)
    float* agg2 = ws + o;

    const long long* esrc = ei;
    const long long* edst = ei + e;

    dim3 B(256);
    int gN   = (n + 255) / 256;
    int gE   = (e + 255) / 256;
    int tot16 = n * H_DIM;
    int g16  = (tot16 + 255) / 256;
    int gE16 = (e * 16 + 255) / 256;
    int ntiles = (n + 15) / 16;
    int gT   = (ntiles + 7) / 8;

    init_k      <<<gN,  B, 0, stream>>>(dinv, sums, n);
    colstats_k  <<<gN,  B, 0, stream>>>(x, sumx, sumsq, n);
    deg_k       <<<gE,  B, 0, stream>>>(edst, dinv, e);
    dinv_k      <<<gN,  B, 0, stream>>>(dinv, n);
    params_k    <<<1,   B, 0, stream>>>(sumx, sumsq, fe, ve, gamma, beta, W1, W2,
                                        Weff, cshift, W2pad, 1.0f / (float)n);
    gemm1_wmma_k<<<gT,  B, 0, stream>>>(x, Weff, cshift, hpre, n);
    agg_init_k  <<<g16, B, 0, stream>>>(dinv, hpre, agg1, tot16);
    scatter_k   <<<gE16,B, 0, stream>>>(esrc, edst, dinv, hpre, agg1, e);
    relu_agg_k  <<<g16, B, 0, stream>>>(agg1, b1, dinv, agg2, tot16);
    scatter_k   <<<gE16,B, 0, stream>>>(esrc, edst, dinv, agg1, agg2, e);
    gemm2_wmma_k<<<gT,  B, 0, stream>>>(agg2, W2pad, b2, out, n);
    logsoftmax_k<<<gN,  B, 0, stream>>>(out, n);
}